// Embedding_31044023616454
// MI455X (gfx1250) — compile-verified
//
#include <hip/hip_runtime.h>
#include <stdint.h>

// Embedding gather: out[r, :] = weight[x[r], :], r in [0, 8192), DIM=128 (512 B/row).
// Pure data movement (0 required FLOPs) -> CDNA5 async global<->LDS copy engine
// (ASYNCcnt), bypassing VGPRs. One wave moves one 512 B row per async op
// (lane l handles bytes [16l,16l+16) -> b128 per lane, fully coalesced).
// GVS addressing: SGPR base (kernarg pointer, already scalar) + 32-bit VGPR
// offset -- weight table is 16 MB, output 4 MB, so offsets fit u32.

#define WAVES_PER_BLOCK 8
#define ROWS_PER_WAVE   4
#define ROWS_PER_BLOCK  (WAVES_PER_BLOCK * ROWS_PER_WAVE)   // 32 rows / block
#define DIM             128
#define ROW_BYTES       (DIM * 4)                            // 512

__global__ __launch_bounds__(WAVES_PER_BLOCK * 32)
void embed_gather_async(const int* __restrict__ x,
                        const float* __restrict__ weight,
                        float* __restrict__ out,
                        int nrows) {
    __shared__ float lds[ROWS_PER_BLOCK * DIM];   // 16 KB per block

    const int lane    = threadIdx.x & 31;
    const int wave    = threadIdx.x >> 5;
    const int rowBase = blockIdx.x * ROWS_PER_BLOCK + wave * ROWS_PER_WAVE;

    if (rowBase + ROWS_PER_WAVE <= nrows) {
        // ---- fast path: full batch of ROWS_PER_WAVE rows via async copies ----
        int idx[ROWS_PER_WAVE];
#pragma unroll
        for (int s = 0; s < ROWS_PER_WAVE; ++s)
            idx[s] = x[rowBase + s];              // lanes share addr -> 1 request

        const uint32_t lane16 = (uint32_t)lane * 16u;

        // LDS byte offset of this wave's slot 0 + this lane's 16B chunk.
        // Low 32 bits of the generic pointer == AS(3) offset, which is what
        // the async ops' LDS-address VGPR expects (HW adds LDS_BASE).
        const uint32_t ldsWaveBase =
            (uint32_t)(uintptr_t)&lds[wave * ROWS_PER_WAVE * DIM] + lane16;

        // Stage 1: gather rows from the table into LDS (async, GVS mode).
#pragma unroll
        for (int s = 0; s < ROWS_PER_WAVE; ++s) {
            uint32_t goff = (uint32_t)idx[s] * (uint32_t)ROW_BYTES + lane16;
            uint32_t la   = ldsWaveBase + (uint32_t)(s * ROW_BYTES);
            asm volatile("global_load_async_to_lds_b128 %0, %1, %2"
                         :: "v"(la), "v"(goff), "s"(weight) : "memory");
        }
        // Loads complete (data visible in LDS) when ASYNCcnt drains.
        asm volatile("s_wait_asynccnt 0x0" ::: "memory");

        // Stage 2: stream rows LDS -> contiguous output (async, GVS, NT store:
        // output is written once, keep it from evicting table rows in L2).
        const uint32_t obase = (uint32_t)rowBase * (uint32_t)ROW_BYTES + lane16;
#pragma unroll
        for (int s = 0; s < ROWS_PER_WAVE; ++s) {
            uint32_t ooff = obase + (uint32_t)(s * ROW_BYTES);
            uint32_t la   = ldsWaveBase + (uint32_t)(s * ROW_BYTES);
            asm volatile("global_store_async_from_lds_b128 %0, %1, %2 th:TH_STORE_NT"
                         :: "v"(ooff), "v"(la), "s"(out) : "memory");
        }
        asm volatile("s_wait_asynccnt 0x0" ::: "memory");
    } else {
        // ---- tail path (not taken for nrows % 32 == 0): plain b128 gather ----
        for (int s = 0; s < ROWS_PER_WAVE; ++s) {
            int r = rowBase + s;
            if (r < nrows) {
                const float4* src = (const float4*)(weight + (size_t)x[r] * DIM);
                float4*       dst = (float4*)(out + (size_t)r * DIM);
                dst[lane] = src[lane];
            }
        }
    }
}

extern "C" void kernel_launch(void* const* d_in, const int* in_sizes, int n_in,
                              void* d_out, int out_size, void* d_ws, size_t ws_size,
                              hipStream_t stream) {
    const int*   x      = (const int*)d_in[0];     // [2,4096] int32 indices
    const float* weight = (const float*)d_in[1];   // [32000,128] f32
    float*       out    = (float*)d_out;           // [2,4096,128] f32

    const int nrows  = in_sizes[0];                // 8192
    const int blocks = (nrows + ROWS_PER_BLOCK - 1) / ROWS_PER_BLOCK;  // 256

    embed_gather_async<<<blocks, WAVES_PER_BLOCK * 32, 0, stream>>>(
        x, weight, out, nrows);
}